// RARL2WithImplicitDiff_10634339025030
// MI455X (gfx1250) — compile-verified
//
#include <hip/hip_runtime.h>
#include <math.h>

typedef float v2f __attribute__((ext_vector_type(2)));
typedef float v8f __attribute__((ext_vector_type(8)));

#define NTHREADS 256
#define NWAVES 8

// Complex GEMM on LDS-resident matrices using V_WMMA_F32_16X16X4_F32.
// D = scale * Op(A) * Op(B) [+ Add],  Op = identity or conjugate-transpose.
// All 16x16 output tiles distributed wave-uniformly (EXEC all-ones per wave).
__device__ __forceinline__ void cgemm(
    const float* Are, const float* Aim, int lda, bool conjA,
    const float* Bre, const float* Bim, int ldb, bool conjB,
    const float* AddRe, const float* AddIm, int ldadd,
    float* Dre, float* Dim, int ldd, float scale,
    int M16, int N16, int K4, int wave, int lane)
{
  const int half = lane >> 4;   // 0 or 1
  const int l15  = lane & 15;
  const int tiles = M16 * N16;
  for (int t = wave; t < tiles; t += NWAVES) {
    const int i16 = (t / N16) << 4;
    const int j16 = (t % N16) << 4;
    v8f rr = {0.f,0.f,0.f,0.f,0.f,0.f,0.f,0.f};
    v8f ii = rr, ri = rr, ir = rr;
#pragma unroll
    for (int k4 = 0; k4 < K4; ++k4) {
      const int k0 = k4 << 2;
      // A fragment: 16x4 block at (i16, k0). Lane l<16: row=l, K=k0,k0+1; l>=16: K=k0+2,k0+3.
      const int arow = i16 + l15;
      const int ak   = k0 + (half << 1);
      v2f are, aim;
      if (!conjA) {
        are.x = Are[arow * lda + ak];      are.y = Are[arow * lda + ak + 1];
        aim.x = Aim[arow * lda + ak];      aim.y = Aim[arow * lda + ak + 1];
      } else {   // Op(A)[r][k] = conj(stored[k][r])
        are.x = Are[ak * lda + arow];      are.y = Are[(ak + 1) * lda + arow];
        aim.x = -Aim[ak * lda + arow];     aim.y = -Aim[(ak + 1) * lda + arow];
      }
      // B fragment: 4x16 block at (k0, j16), symmetric lane split in K.
      const int bcol = j16 + l15;
      const int bk   = k0 + (half << 1);
      v2f bre, bim;
      if (!conjB) {
        bre.x = Bre[bk * ldb + bcol];      bre.y = Bre[(bk + 1) * ldb + bcol];
        bim.x = Bim[bk * ldb + bcol];      bim.y = Bim[(bk + 1) * ldb + bcol];
      } else {   // Op(B)[k][c] = conj(stored[c][k])
        bre.x = Bre[bcol * ldb + bk];      bre.y = Bre[bcol * ldb + bk + 1];
        bim.x = -Bim[bcol * ldb + bk];     bim.y = -Bim[bcol * ldb + bk + 1];
      }
      // F32 WMMA has no A/B negate (NEG = CNeg only) -> 4 positive chains,
      // combine with VALU below.
      rr = __builtin_amdgcn_wmma_f32_16x16x4_f32(false, are, false, bre, (short)0, rr, false, false);
      ii = __builtin_amdgcn_wmma_f32_16x16x4_f32(false, aim, false, bim, (short)0, ii, false, false);
      ri = __builtin_amdgcn_wmma_f32_16x16x4_f32(false, are, false, bim, (short)0, ri, false, false);
      ir = __builtin_amdgcn_wmma_f32_16x16x4_f32(false, aim, false, bre, (short)0, ir, false, false);
    }
    // C/D layout: VGPR r -> row (i16 + half*8 + r), col j16 + l15.
    const int col  = j16 + l15;
    const int rowb = i16 + (half << 3);
#pragma unroll
    for (int r = 0; r < 8; ++r) {
      float dr = scale * (rr[r] - ii[r]);
      float di = scale * (ri[r] + ir[r]);
      if (AddRe) {
        dr += AddRe[(rowb + r) * ldadd + col];
        di += AddIm[(rowb + r) * ldadd + col];
      }
      Dre[(rowb + r) * ldd + col] = dr;
      Dim[(rowb + r) * ldd + col] = di;
    }
  }
}

// LDS layout (floats). Five 64x64 complex "big" buffers + small panels.
#define OFF_AF_RE 0
#define OFF_AF_IM 4096
#define OFF_A_RE  8192
#define OFF_A_IM  12288
#define OFF_R_RE  16384
#define OFF_R_IM  20480
#define OFF_X_RE  24576
#define OFF_X_IM  28672
#define OFF_T_RE  32768
#define OFF_T_IM  36864
#define OFF_V_RE  40960
#define OFF_V_IM  41984
#define OFF_CF_RE 43008
#define OFF_CF_IM 44032
#define OFF_C_RE  45056
#define OFF_C_IM  46080
#define OFF_CH_RE 47104
#define OFF_CH_IM 48128
#define OFF_BF_RE 49152
#define OFF_BF_IM 50176
#define OFF_BH_RE 51200
#define OFF_BH_IM 52224
#define OFF_W_RE  53248
#define OFF_W_IM  54272
#define OFF_W2_RE 55296
#define OFF_W2_IM 56320
#define OFF_DL_RE 57344
#define OFF_DL_IM 57600
#define OFF_DF_RE 57856
#define OFF_DF_IM 58112
#define OFF_RED   58368
#define SMEM_FLOATS 58376

__global__ void __launch_bounds__(NTHREADS)
rarl2_kernel(const double* __restrict__ gVr, const double* __restrict__ gVi,
             const double* __restrict__ gAF, const double* __restrict__ gBF,
             const double* __restrict__ gCF, const double* __restrict__ gDF,
             float* __restrict__ out)
{
  extern __shared__ float sm[];
  const int tid  = threadIdx.x;
  const int wave = tid >> 5;
  const int lane = tid & 31;

  float* AFre = sm + OFF_AF_RE; float* AFim = sm + OFF_AF_IM;
  float* Are  = sm + OFF_A_RE;  float* Aim  = sm + OFF_A_IM;
  float* Rre  = sm + OFF_R_RE;  float* Rim  = sm + OFF_R_IM;   // RHS buffer
  float* Xre  = sm + OFF_X_RE;  float* Xim  = sm + OFF_X_IM;
  float* Tre  = sm + OFF_T_RE;  float* Tim  = sm + OFF_T_IM;   // temp / Cholesky R
  float* Vre  = sm + OFF_V_RE;  float* Vim  = sm + OFF_V_IM;   // 16x64
  float* CFre = sm + OFF_CF_RE; float* CFim = sm + OFF_CF_IM;  // 16x64
  float* Cre  = sm + OFF_C_RE;  float* Cim  = sm + OFF_C_IM;   // 16x64
  float* CHre = sm + OFF_CH_RE; float* CHim = sm + OFF_CH_IM;  // 64x16
  float* BFre = sm + OFF_BF_RE; float* BFim = sm + OFF_BF_IM;  // 64x16
  float* BHre = sm + OFF_BH_RE; float* BHim = sm + OFF_BH_IM;  // 64x16
  float* Wre  = sm + OFF_W_RE;  float* Wim  = sm + OFF_W_IM;   // 64x16
  float* W2re = sm + OFF_W2_RE; float* W2im = sm + OFF_W2_IM;  // 64x16
  float* DLre = sm + OFF_DL_RE; float* DLim = sm + OFF_DL_IM;  // 16x16
  float* DFre = sm + OFF_DF_RE; float* DFim = sm + OFF_DF_IM;  // 16x16
  float* red  = sm + OFF_RED;

  // ---- zero all LDS (padding correctness depends on it) ----
  for (int i = tid; i < SMEM_FLOATS; i += NTHREADS) sm[i] = 0.0f;
  __syncthreads();

  // ---- load + f64->f32 convert + pad ----
  for (int i = tid; i < 8 * 48; i += NTHREADS) {          // V: 8x48 real pair
    int r = i / 48, c = i % 48;
    Vre[r * 64 + c] = (float)gVr[i];
    Vim[r * 64 + c] = (float)gVi[i];
  }
  for (int i = tid; i < 64 * 64; i += NTHREADS) {         // A_F: complex128
    AFre[i] = (float)gAF[2 * i];
    AFim[i] = (float)gAF[2 * i + 1];
  }
  for (int i = tid; i < 64 * 8; i += NTHREADS) {          // B_F: 64x8
    int r = i / 8, c = i % 8;
    BFre[r * 16 + c] = (float)gBF[2 * i];
    BFim[r * 16 + c] = (float)gBF[2 * i + 1];
  }
  for (int i = tid; i < 8 * 64; i += NTHREADS) {          // C_F: 8x64
    int r = i / 64, c = i % 64;
    CFre[r * 64 + c] = (float)gCF[2 * i];
    CFim[r * 64 + c] = (float)gCF[2 * i + 1];
  }
  for (int i = tid; i < 8 * 8; i += NTHREADS) {           // D_F: 8x8
    int r = i / 8, c = i % 8;
    DFre[r * 16 + c] = (float)gDF[2 * i];
    DFim[r * 16 + c] = (float)gDF[2 * i + 1];
  }
  __syncthreads();

  // ---- Cholesky-QR of [V; I]: G = I + V^H V (into T), G = R^H R ----
  cgemm(Vre, Vim, 64, true,  Vre, Vim, 64, false,
        nullptr, nullptr, 0, Tre, Tim, 64, 1.0f, 4, 4, 4, wave, lane);
  __syncthreads();
  for (int k = tid; k < 48; k += NTHREADS) Tre[k * 64 + k] += 1.0f;
  __syncthreads();

  for (int k = 0; k < 48; ++k) {
    if (tid == 0) {
      float d = sqrtf(fmaxf(Tre[k * 64 + k], 1e-30f));
      Tre[k * 64 + k] = d;
      Tim[k * 64 + k] = 0.0f;
      red[4] = 1.0f / d;
    }
    __syncthreads();
    const float invd = red[4];
    for (int j = k + 1 + tid; j < 48; j += NTHREADS) {
      Tre[k * 64 + j] *= invd;
      Tim[k * 64 + j] *= invd;
    }
    __syncthreads();
    const int rem = 47 - k;
    for (int p = tid; p < rem * rem; p += NTHREADS) {
      int i = k + 1 + p / rem;
      int j = k + 1 + p % rem;
      if (j >= i) {       // G[i][j] -= conj(R[k][i]) * R[k][j]
        float ar = Tre[k * 64 + i], ai = -Tim[k * 64 + i];
        float br = Tre[k * 64 + j], bi =  Tim[k * 64 + j];
        Tre[i * 64 + j] -= ar * br - ai * bi;
        Tim[i * 64 + j] -= ar * bi + ai * br;
      }
    }
    __syncthreads();
  }

  // ---- A = R^{-1} (upper-triangular inverse, one column per thread) ----
  if (tid < 48) {
    const int j = tid;
    Are[j * 64 + j] = 1.0f / Tre[j * 64 + j];
    Aim[j * 64 + j] = 0.0f;
    for (int i = j - 1; i >= 0; --i) {
      float sr = 0.f, si = 0.f;
      for (int t2 = i + 1; t2 <= j; ++t2) {
        float rr2 = Tre[i * 64 + t2], ri2 = Tim[i * 64 + t2];
        float xr2 = Are[t2 * 64 + j], xi2 = Aim[t2 * 64 + j];
        sr += rr2 * xr2 - ri2 * xi2;
        si += rr2 * xi2 + ri2 * xr2;
      }
      float inv = 1.0f / Tre[i * 64 + i];
      Are[i * 64 + j] = -sr * inv;
      Aim[i * 64 + j] = -si * inv;
    }
  }
  __syncthreads();

  // ---- C = V * A ----
  cgemm(Vre, Vim, 64, false, Are, Aim, 64, false,
        nullptr, nullptr, 0, Cre, Cim, 64, 1.0f, 1, 4, 16, wave, lane);
  __syncthreads();
  // CH = C^H  and  RHS = C_F^H * C
  for (int i = tid; i < 64 * 16; i += NTHREADS) {
    int r = i / 16, c = i % 16;
    CHre[i] =  Cre[c * 64 + r];
    CHim[i] = -Cim[c * 64 + r];
  }
  cgemm(CFre, CFim, 64, true, Cre, Cim, 64, false,
        nullptr, nullptr, 0, Rre, Rim, 64, 1.0f, 4, 4, 4, wave, lane);
  __syncthreads();

  // ---- Stein fixed point: X <- A_F^H X A + RHS (X zeroed already) ----
#pragma unroll 1
  for (int it = 0; it < 50; ++it) {
    cgemm(AFre, AFim, 64, true, Xre, Xim, 64, false,
          nullptr, nullptr, 0, Tre, Tim, 64, 1.0f, 4, 4, 16, wave, lane);
    __syncthreads();
    cgemm(Tre, Tim, 64, false, Are, Aim, 64, false,
          Rre, Rim, 64, Xre, Xim, 64, 1.0f, 4, 4, 16, wave, lane);
    __syncthreads();
  }
  // B_hat = -X^H B_F  (snapshot at 50 iterations, as in the reference)
  cgemm(Xre, Xim, 64, true, BFre, BFim, 16, false,
        nullptr, nullptr, 0, BHre, BHim, 16, -1.0f, 4, 1, 16, wave, lane);
  __syncthreads();

  // ---- continue to convergence (replaces 3072x3072 Kronecker solve) ----
#pragma unroll 1
  for (int it = 0; it < 250; ++it) {
    cgemm(AFre, AFim, 64, true, Xre, Xim, 64, false,
          nullptr, nullptr, 0, Tre, Tim, 64, 1.0f, 4, 4, 16, wave, lane);
    __syncthreads();
    cgemm(Tre, Tim, 64, false, Are, Aim, 64, false,
          Rre, Rim, 64, Xre, Xim, 64, 1.0f, 4, 4, 16, wave, lane);
    __syncthreads();
  }
  // reduction = || X^H B_F ||_F^2
  cgemm(Xre, Xim, 64, true, BFre, BFim, 16, false,
        nullptr, nullptr, 0, Wre, Wim, 16, 1.0f, 4, 1, 16, wave, lane);
  __syncthreads();
  {
    float part = 0.f;
    for (int i = tid; i < 64 * 16; i += NTHREADS)
      part += Wre[i] * Wre[i] + Wim[i] * Wim[i];
    atomicAdd(&red[0], part);
  }
  __syncthreads();

  // ---- ||F||_H2^2: Qg <- A_F^H Qg A_F + C_F^H C_F (replaces 4096^2 dlyap) ----
  cgemm(CFre, CFim, 64, true, CFre, CFim, 64, false,
        nullptr, nullptr, 0, Rre, Rim, 64, 1.0f, 4, 4, 4, wave, lane);
  for (int i = tid; i < 4096; i += NTHREADS) { Xre[i] = 0.f; Xim[i] = 0.f; }
  __syncthreads();
#pragma unroll 1
  for (int it = 0; it < 240; ++it) {
    cgemm(AFre, AFim, 64, true, Xre, Xim, 64, false,
          nullptr, nullptr, 0, Tre, Tim, 64, 1.0f, 4, 4, 16, wave, lane);
    __syncthreads();
    cgemm(Tre, Tim, 64, false, AFre, AFim, 64, false,
          Rre, Rim, 64, Xre, Xim, 64, 1.0f, 4, 4, 16, wave, lane);
    __syncthreads();
  }
  // trace(B_F^H Qg B_F).re + sum|D_F|^2
  cgemm(Xre, Xim, 64, false, BFre, BFim, 16, false,
        nullptr, nullptr, 0, W2re, W2im, 16, 1.0f, 4, 1, 16, wave, lane);
  __syncthreads();
  {
    float tpart = 0.f;
    for (int i = tid; i < 64 * 16; i += NTHREADS)
      tpart += BFre[i] * W2re[i] + BFim[i] * W2im[i];   // Re(conj(BF)*U)
    for (int i = tid; i < 16 * 16; i += NTHREADS)
      tpart += DFre[i] * DFre[i] + DFim[i] * DFim[i];
    atomicAdd(&red[1], tpart);
  }
  __syncthreads();

  // ---- D_hat: W = (I - A^H)^{-1} C^H via Neumann series (ping-pong) ----
  for (int i = tid; i < 1024; i += NTHREADS) { Wre[i] = 0.f; Wim[i] = 0.f; }
  __syncthreads();
#pragma unroll 1
  for (int it = 0; it < 128; ++it) {
    cgemm(Are, Aim, 64, true, Wre, Wim, 16, false,
          CHre, CHim, 16, W2re, W2im, 16, 1.0f, 4, 1, 16, wave, lane);
    __syncthreads();
    cgemm(Are, Aim, 64, true, W2re, W2im, 16, false,
          CHre, CHim, 16, Wre, Wim, 16, 1.0f, 4, 1, 16, wave, lane);
    __syncthreads();
  }
  // D_lossless = I_8 - C * W
  cgemm(Cre, Cim, 64, false, Wre, Wim, 16, false,
        nullptr, nullptr, 0, DLre, DLim, 16, -1.0f, 1, 1, 16, wave, lane);
  __syncthreads();
  if (tid < 8) DLre[tid * 16 + tid] += 1.0f;
  __syncthreads();

  // ---- outputs: [loss, B_hat(48x8 cplx), D_hat(8x8 cplx)] as float ----
  if (tid == 0) {
    float loss = red[1] - red[0];
    out[0] = loss > 0.f ? loss : 0.f;
  }
  for (int i = tid; i < 48 * 8; i += NTHREADS) {
    int r = i / 8, c = i % 8;
    out[1 + 2 * i]     = BHre[r * 16 + c];
    out[1 + 2 * i + 1] = BHim[r * 16 + c];
  }
  for (int i = tid; i < 64; i += NTHREADS) {
    int r = i / 8, c = i % 8;
    float sr = 0.f, si = 0.f;
#pragma unroll
    for (int k = 0; k < 8; ++k) { sr += DLre[r * 16 + k]; si += DLim[r * 16 + k]; }
    // 0.5 * (D_lossless @ ones)/P + 0.5 * D_F ; 0.5/8 = 0.0625
    out[1 + 768 + 2 * i]     = 0.0625f * sr + 0.5f * DFre[r * 16 + c];
    out[1 + 768 + 2 * i + 1] = 0.0625f * si + 0.5f * DFim[r * 16 + c];
  }
}

extern "C" void kernel_launch(void* const* d_in, const int* in_sizes, int n_in,
                              void* d_out, int out_size, void* d_ws, size_t ws_size,
                              hipStream_t stream) {
  (void)in_sizes; (void)n_in; (void)d_ws; (void)ws_size; (void)out_size;
  const double* Vr = (const double*)d_in[0];
  const double* Vi = (const double*)d_in[1];
  const double* AF = (const double*)d_in[2];
  const double* BF = (const double*)d_in[3];
  const double* CF = (const double*)d_in[4];
  const double* DF = (const double*)d_in[5];
  const size_t shmem = (size_t)SMEM_FLOATS * sizeof(float);
  rarl2_kernel<<<dim3(1), dim3(NTHREADS), shmem, stream>>>(
      Vr, Vi, AF, BF, CF, DF, (float*)d_out);
}